// Cluster_Pool__47296179863968
// MI455X (gfx1250) — compile-verified
//
#include <hip/hip_runtime.h>

#define BN       32768
#define NB       4096
#define NBATCH   8
#define MCLU     16
#define CDIM     67
#define DDIM     3
#define FDIM     64
#define E_EDGES  524288
#define KM_ITERS 20
#define KSPLIT   64
#define CHUNK    (NB / KSPLIT)   // 64 points per K-split wave

typedef float v2f __attribute__((ext_vector_type(2)));
typedef float v8f __attribute__((ext_vector_type(8)));

// ---------------- Phase 0: per-batch KMeans entirely in LDS ----------------
__global__ void kmeans_kernel(const float* __restrict__ x, float* __restrict__ cent) {
  __shared__ float spos[NB * 3];       // 48 KB: all positions of this batch
  __shared__ float scent[MCLU * 3];
  __shared__ float ssum[MCLU * 4];     // xsum, ysum, zsum, count
  const int b = blockIdx.x;
  const int tid = threadIdx.x;
  const float* xb = x + (size_t)b * NB * CDIM;
  for (int i = tid; i < NB; i += blockDim.x) {
    spos[i * 3 + 0] = xb[(size_t)i * CDIM + 0];
    spos[i * 3 + 1] = xb[(size_t)i * CDIM + 1];
    spos[i * 3 + 2] = xb[(size_t)i * CDIM + 2];
  }
  __syncthreads();
  if (tid < MCLU * 3) scent[tid] = spos[tid];   // deterministic init: first 16 pts
  __syncthreads();
  for (int it = 0; it < KM_ITERS; ++it) {
    if (tid < MCLU * 4) ssum[tid] = 0.0f;
    __syncthreads();
    for (int i = tid; i < NB; i += blockDim.x) {
      const float px = spos[i * 3 + 0], py = spos[i * 3 + 1], pz = spos[i * 3 + 2];
      int best = 0; float bestd = 3.4e38f;
      #pragma unroll
      for (int m = 0; m < MCLU; ++m) {
        const float dx = px - scent[m * 3 + 0];
        const float dy = py - scent[m * 3 + 1];
        const float dz = pz - scent[m * 3 + 2];
        const float d = dx * dx + dy * dy + dz * dz;
        if (d < bestd) { bestd = d; best = m; }   // argmin: first min wins (ties)
      }
      atomicAdd(&ssum[best * 4 + 0], px);
      atomicAdd(&ssum[best * 4 + 1], py);
      atomicAdd(&ssum[best * 4 + 2], pz);
      atomicAdd(&ssum[best * 4 + 3], 1.0f);
    }
    __syncthreads();
    if (tid < MCLU) {
      const float cnt = ssum[tid * 4 + 3];
      if (cnt > 0.0f) {
        scent[tid * 3 + 0] = ssum[tid * 4 + 0] / cnt;
        scent[tid * 3 + 1] = ssum[tid * 4 + 1] / cnt;
        scent[tid * 3 + 2] = ssum[tid * 4 + 2] / cnt;
      }
    }
    __syncthreads();
  }
  if (tid < MCLU * 3) cent[b * MCLU * 3 + tid] = scent[tid];
}

// ---------------- Phase 1: S = softmax(-dist^2) ----------------
__global__ void assign_kernel(const float* __restrict__ x, const float* __restrict__ cent,
                              float* __restrict__ S) {
  __shared__ float scent[MCLU * 3];
  const int n = blockIdx.x * blockDim.x + threadIdx.x;   // block spans one batch only
  const int b = n >> 12;
  if (threadIdx.x < MCLU * 3) scent[threadIdx.x] = cent[b * MCLU * 3 + threadIdx.x];
  __syncthreads();
  const float px = x[(size_t)n * CDIM + 0];
  const float py = x[(size_t)n * CDIM + 1];
  const float pz = x[(size_t)n * CDIM + 2];
  float d[MCLU]; float dmin = 3.4e38f;
  #pragma unroll
  for (int m = 0; m < MCLU; ++m) {
    const float dx = px - scent[m * 3 + 0];
    const float dy = py - scent[m * 3 + 1];
    const float dz = pz - scent[m * 3 + 2];
    d[m] = dx * dx + dy * dy + dz * dz;
    dmin = fminf(dmin, d[m]);
  }
  float e[MCLU]; float sum = 0.0f;
  #pragma unroll
  for (int m = 0; m < MCLU; ++m) { e[m] = expf(dmin - d[m]); sum += e[m]; }
  const float inv = 1.0f / sum;
  #pragma unroll
  for (int m = 0; m < MCLU; ++m) S[(size_t)n * MCLU + m] = e[m] * inv;
}

// ---------------- zero fill ----------------
__global__ void zero_kernel(float* __restrict__ p, int n) {
  const int i = blockIdx.x * blockDim.x + threadIdx.x;
  if (i < n) p[i] = 0.0f;
}

// ---------------- Phase 2: AS[e0] += S[e1]  (dominant memory phase) ----------------
__global__ void scatter_kernel(const int* __restrict__ e, const float* __restrict__ S,
                               float* __restrict__ AS) {
  const int idx = blockIdx.x * blockDim.x + threadIdx.x;   // E*16 threads exactly
  const int j = idx >> 4;
  const int k = idx & 15;
  const int e0 = e[j];
  const int e1 = e[E_EDGES + j];
  atomicAdd(&AS[(size_t)e0 * MCLU + k], S[(size_t)e1 * MCLU + k]);
}

// ---------------- Phase 3: WMMA f32 16x16x4 GEMMs ----------------
// tile 0..3: pooled[b] (16x64) = S_b^T (16xN) * F_b (Nx64), column tile of 16
// tile 4   : A_MM[b]  (16x16) = S_b^T (16xN) * AS_b (Nx16)
// B-source selection is hoisted to a uniform (pointer, stride) pair so the
// inner loop is branch-free: 4 loads + 1 v_wmma per K-step.
__global__ void wmma_gemm_kernel(const float* __restrict__ x, const float* __restrict__ S,
                                 const float* __restrict__ AS, float* __restrict__ pooled,
                                 float* __restrict__ amm) {
  int blk = blockIdx.x;
  const int ks = blk % KSPLIT; blk /= KSPLIT;
  const int tile = blk % 5;
  const int b = blk / 5;
  const int lane = threadIdx.x;        // one wave32 per block
  const int half = lane >> 4;          // lanes 16-31 hold K+2 / rows M+8
  const int l16 = lane & 15;
  const int n0 = b * NB + ks * CHUNK + 2 * half;

  // Uniform B-matrix source: (base, row-stride) chosen once per block.
  const float* bsrc;
  size_t bstride;
  if (tile < 4) { bsrc = x + DDIM + (size_t)tile * 16; bstride = CDIM; }
  else          { bsrc = AS;                           bstride = MCLU; }

  // Per-lane streaming pointers (A layout: v0=K(0|2), v1=K(1|3); M = lane&15)
  const float* aptr = S + (size_t)n0 * MCLU + l16;
  const float* bptr = bsrc + (size_t)n0 * bstride + l16;
  const size_t astep = 4 * (size_t)MCLU;
  const size_t bstep = 4 * bstride;

  v8f acc = {0.f, 0.f, 0.f, 0.f, 0.f, 0.f, 0.f, 0.f};
  #pragma unroll 4
  for (int kk = 0; kk < CHUNK; kk += 4) {
    v2f a, bm;
    a.x  = aptr[0];
    a.y  = aptr[MCLU];
    bm.x = bptr[0];
    bm.y = bptr[bstride];
    aptr += astep;
    bptr += bstep;
    acc = __builtin_amdgcn_wmma_f32_16x16x4_f32(
        /*neg_a=*/false, a, /*neg_b=*/false, bm,
        /*c_mod=*/(short)0, acc, /*reuse_a=*/false, /*reuse_b=*/false);
  }
  // D layout: VGPR i -> row (i + 8*half), col l16
  #pragma unroll
  for (int i = 0; i < 8; ++i) {
    const int m = i + 8 * half;
    if (tile < 4) atomicAdd(&pooled[((size_t)b * MCLU + m) * FDIM + tile * 16 + l16], acc[i]);
    else          atomicAdd(&amm[((size_t)b * MCLU + m) * MCLU + l16], acc[i]);
  }
}

// ---------------- Phase 4a: pack x_out = [centroids | pooled] ----------------
__global__ void pack_kernel(const float* __restrict__ cent, const float* __restrict__ pooled,
                            float* __restrict__ out) {
  const int idx = blockIdx.x * blockDim.x + threadIdx.x;
  if (idx >= 128 * CDIM) return;
  const int row = idx / CDIM;
  const int col = idx % CDIM;
  out[idx] = (col < DDIM) ? cent[row * 3 + col] : pooled[row * FDIM + (col - DDIM)];
}

// ---------------- Phase 4b: stable top-4 per A_MM row -> e_out, b_out ----------------
__global__ void topk_kernel(const float* __restrict__ amm, float* __restrict__ out) {
  const int idx = threadIdx.x;           // 128 threads: (b, m)
  const int b = idx >> 4;
  const int m = idx & 15;
  const float* row = amm + ((size_t)b * MCLU + m) * MCLU;
  float v[MCLU];
  #pragma unroll
  for (int k = 0; k < MCLU; ++k) v[k] = row[k];
  unsigned used = 0;
  const int XO = 128 * CDIM;             // 8576
  #pragma unroll
  for (int t = 0; t < 4; ++t) {
    float bv = -3.4e38f; int bi = 0;
    #pragma unroll
    for (int k = 0; k < MCLU; ++k) {
      if (!((used >> k) & 1u) && v[k] > bv) { bv = v[k]; bi = k; }  // ties: lowest idx
    }
    used |= 1u << bi;
    out[XO + idx * 4 + t]       = (float)(m + b * MCLU);   // e_out row 0 (src)
    out[XO + 512 + idx * 4 + t] = (float)(bi + b * MCLU);  // e_out row 1 (dst)
  }
  out[XO + 1024 + idx] = (float)b;                         // b_out
}

extern "C" void kernel_launch(void* const* d_in, const int* in_sizes, int n_in,
                              void* d_out, int out_size, void* d_ws, size_t ws_size,
                              hipStream_t stream) {
  const float* x = (const float*)d_in[0];
  const int*   e = (const int*)d_in[1];
  float* ws = (float*)d_ws;
  float* cent   = ws;                            // 384
  float* S      = cent + NBATCH * MCLU * 3;      // 524288
  float* AS     = S + (size_t)BN * MCLU;         // 524288
  float* pooled = AS + (size_t)BN * MCLU;        // 8192
  float* amm    = pooled + NBATCH * MCLU * FDIM; // 2048
  float* out = (float*)d_out;

  kmeans_kernel<<<NBATCH, 512, 0, stream>>>(x, cent);
  assign_kernel<<<BN / 256, 256, 0, stream>>>(x, cent, S);
  const int zn = BN * MCLU + NBATCH * MCLU * FDIM + NBATCH * MCLU * MCLU;
  zero_kernel<<<(zn + 255) / 256, 256, 0, stream>>>(AS, zn);
  scatter_kernel<<<(E_EDGES * 16) / 256, 256, 0, stream>>>(e, S, AS);
  wmma_gemm_kernel<<<NBATCH * 5 * KSPLIT, 32, 0, stream>>>(x, S, AS, pooled, amm);
  pack_kernel<<<(128 * CDIM + 255) / 256, 256, 0, stream>>>(cent, pooled, out);
  topk_kernel<<<1, 128, 0, stream>>>(amm, out);
}